// GAT_61289183314543
// MI455X (gfx1250) — compile-verified
//
#include <hip/hip_runtime.h>
#include <math.h>

#define DFEAT 64
#define NHEAD 4
#define HD (NHEAD * DFEAT)   // 256
#define RB 200               // reduction blocks

typedef float v2f __attribute__((ext_vector_type(2)));
typedef float v8f __attribute__((ext_vector_type(8)));

__device__ __forceinline__ float eluf(float x) { return x > 0.f ? x : (expf(x) - 1.f); }
__device__ __forceinline__ float4 f4max(float4 a, float4 b) {
    return make_float4(fmaxf(a.x, b.x), fmaxf(a.y, b.y), fmaxf(a.z, b.z), fmaxf(a.w, b.w));
}
__device__ __forceinline__ float4 f4add(float4 a, float4 b) {
    return make_float4(a.x + b.x, a.y + b.y, a.z + b.z, a.w + b.w);
}

// ---------------- K0: zero the neigh accumulator ----------------
__global__ void k_zero4(float4* p, long n4) {
    long i = (long)blockIdx.x * blockDim.x + threadIdx.x;
    long stride = (long)gridDim.x * blockDim.x;
    float4 z = make_float4(0.f, 0.f, 0.f, 0.f);
    for (; i < n4; i += stride) p[i] = z;
}

// ---------------- K1: scores[n,h] = (h_n . Wh_h . h_n) / 8 ----------------
// block = 256 threads -> 64 nodes x 4 heads. Wh staged in LDS as [d][e][h]
// so lanes differing only in h hit 4 consecutive banks (broadcast across nodes).
__global__ void k_scores(const float* __restrict__ x, const float* __restrict__ Wh,
                         float* __restrict__ scores, int nnodes) {
    __shared__ float Wl[DFEAT * DFEAT * NHEAD];   // 64 KB
    for (int idx = threadIdx.x; idx < DFEAT * DFEAT * NHEAD; idx += blockDim.x) {
        int h = idx >> 12;            // / (64*64)
        int rem = idx & 4095;
        int d = rem >> 6, e = rem & 63;
        Wl[(d << 8) + (e << 2) + h] = Wh[idx];
    }
    __syncthreads();
    int t = threadIdx.x;
    int node = blockIdx.x * 64 + (t >> 2);
    int hh = t & 3;
    if (node >= nnodes) return;

    float hreg[DFEAT];
    const float4* xr = (const float4*)(x + (size_t)node * DFEAT);
#pragma unroll
    for (int i = 0; i < 16; ++i) {
        float4 v = xr[i];
        hreg[4 * i] = v.x; hreg[4 * i + 1] = v.y; hreg[4 * i + 2] = v.z; hreg[4 * i + 3] = v.w;
    }
    float s = 0.f;
    for (int e = 0; e < DFEAT; ++e) {
        float tacc = 0.f;
#pragma unroll 8
        for (int d = 0; d < DFEAT; ++d)
            tacc = fmaf(hreg[d], Wl[(d << 8) + (e << 2) + hh], tacc);
        s = fmaf(tacc, hreg[e], s);
    }
    scores[(size_t)node * NHEAD + hh] = s * 0.125f;   // 1/sqrt(64)
}

// ---------------- K2: per-head partial max over nodes ----------------
__global__ void k_pmax(const float4* __restrict__ sc, float4* __restrict__ partial, int n) {
    __shared__ float4 red[256];
    float4 m = make_float4(-3.0e38f, -3.0e38f, -3.0e38f, -3.0e38f);
    for (int i = blockIdx.x * blockDim.x + threadIdx.x; i < n; i += gridDim.x * blockDim.x)
        m = f4max(m, sc[i]);
    red[threadIdx.x] = m;
    __syncthreads();
    for (int s = 128; s > 0; s >>= 1) {
        if (threadIdx.x < s) red[threadIdx.x] = f4max(red[threadIdx.x], red[threadIdx.x + s]);
        __syncthreads();
    }
    if (threadIdx.x == 0) partial[blockIdx.x] = red[0];
}

__global__ void k_fmax(const float4* __restrict__ partial, float4* __restrict__ stats, int nb) {
    __shared__ float4 red[256];
    float4 m = make_float4(-3.0e38f, -3.0e38f, -3.0e38f, -3.0e38f);
    if ((int)threadIdx.x < nb) m = partial[threadIdx.x];
    red[threadIdx.x] = m;
    __syncthreads();
    for (int s = 128; s > 0; s >>= 1) {
        if (threadIdx.x < s) red[threadIdx.x] = f4max(red[threadIdx.x], red[threadIdx.x + s]);
        __syncthreads();
    }
    if (threadIdx.x == 0) stats[0] = red[0];
}

// ---------------- K3: per-head partial sum of exp(score - max) ----------------
__global__ void k_psum(const float4* __restrict__ sc, const float4* __restrict__ stats,
                       float4* __restrict__ partial, int n) {
    __shared__ float4 red[256];
    float4 mx = stats[0];
    float4 s = make_float4(0.f, 0.f, 0.f, 0.f);
    for (int i = blockIdx.x * blockDim.x + threadIdx.x; i < n; i += gridDim.x * blockDim.x) {
        float4 v = sc[i];
        s.x += expf(v.x - mx.x); s.y += expf(v.y - mx.y);
        s.z += expf(v.z - mx.z); s.w += expf(v.w - mx.w);
    }
    red[threadIdx.x] = s;
    __syncthreads();
    for (int st = 128; st > 0; st >>= 1) {
        if (threadIdx.x < st) red[threadIdx.x] = f4add(red[threadIdx.x], red[threadIdx.x + st]);
        __syncthreads();
    }
    if (threadIdx.x == 0) partial[blockIdx.x] = red[0];
}

__global__ void k_fsum(const float4* __restrict__ partial, float4* __restrict__ stats, int nb) {
    __shared__ float4 red[256];
    float4 s = make_float4(0.f, 0.f, 0.f, 0.f);
    if ((int)threadIdx.x < nb) s = partial[threadIdx.x];
    red[threadIdx.x] = s;
    __syncthreads();
    for (int st = 128; st > 0; st >>= 1) {
        if (threadIdx.x < st) red[threadIdx.x] = f4add(red[threadIdx.x], red[threadIdx.x + st]);
        __syncthreads();
    }
    if (threadIdx.x == 0) stats[1] = red[0];
}

// ---------------- K4: attn = exp(score - max) / sum   (in place) ----------------
__global__ void k_attn(float4* __restrict__ sc, const float4* __restrict__ stats, int n) {
    float4 mx = stats[0];
    float4 sm = stats[1];
    float4 r = make_float4(1.f / sm.x, 1.f / sm.y, 1.f / sm.z, 1.f / sm.w);
    for (int i = blockIdx.x * blockDim.x + threadIdx.x; i < n; i += gridDim.x * blockDim.x) {
        float4 v = sc[i];
        sc[i] = make_float4(expf(v.x - mx.x) * r.x, expf(v.y - mx.y) * r.y,
                            expf(v.z - mx.z) * r.z, expf(v.w - mx.w) * r.w);
    }
}

// ---------------- K5: edge scatter  neigh[dst] += h[src] * attn[src] ----------------
// 256 threads = one (h,d) element each; 32 edges per block.
__global__ void k_edges(const float* __restrict__ x, const float* __restrict__ attn,
                        const int* __restrict__ src, const int* __restrict__ dst,
                        float* __restrict__ neigh, int E) {
    int t = threadIdx.x;
    int d = t & 63;
    int hh = t >> 6;
    int e0 = blockIdx.x * 32;
    int eend = min(e0 + 32, E);
    for (int e = e0; e < eend; ++e) {
        int s = src[e];
        int v = dst[e];
        float val = x[(size_t)s * DFEAT + d] * attn[(size_t)s * NHEAD + hh];
        unsafeAtomicAdd(&neigh[(size_t)v * HD + t], val);   // global_atomic_add_f32, no return
    }
}

// ---------------- K6: out = ELU(neigh[N,256] @ W[256,64]) via V_WMMA_F32_16X16X4_F32 ----
// blockDim = 128 (4 waves). Block handles 16 rows; wave w owns cols [16w,16w+16).
__global__ void k_out(const float* __restrict__ neigh, const float* __restrict__ W,
                      float* __restrict__ out, int n) {
    __shared__ float As[16 * HD];   // 16 KB A tile
    int tid = threadIdx.x;
    int base = blockIdx.x * 16;
    size_t lim = (size_t)n * HD;
    for (int idx = tid; idx < 16 * HD; idx += 128) {
        size_t g = (size_t)base * HD + idx;
        As[idx] = (g < lim) ? neigh[g] : 0.f;
    }
    __syncthreads();

    int lane = tid & 31;
    int wv = tid >> 5;                 // 0..3 -> column tile
    int col = (wv << 4) + (lane & 15);
    int hi = lane >> 4;                // 0: K pair {0,1}; 1: K pair {2,3}
    int arow = lane & 15;

    v8f acc = {0.f, 0.f, 0.f, 0.f, 0.f, 0.f, 0.f, 0.f};
    for (int kk = 0; kk < HD; kk += 4) {
        int ak = kk + (hi << 1);
        v2f a, b;
        a.x = As[arow * HD + ak];
        a.y = As[arow * HD + ak + 1];
        b.x = W[(size_t)ak * DFEAT + col];
        b.y = W[(size_t)(ak + 1) * DFEAT + col];
        // 8 args: (neg_a, A, neg_b, B, c_mod, C, reuse_a, reuse_b)
        acc = __builtin_amdgcn_wmma_f32_16x16x4_f32(false, a, false, b, (short)0, acc,
                                                    false, false);
    }
#pragma unroll
    for (int j = 0; j < 8; ++j) {
        int row = base + j + (hi << 3);
        if (row < n) out[(size_t)row * DFEAT + col] = eluf(acc[j]);
    }
}

extern "C" void kernel_launch(void* const* d_in, const int* in_sizes, int n_in,
                              void* d_out, int out_size, void* d_ws, size_t ws_size,
                              hipStream_t stream) {
    const float* x   = (const float*)d_in[0];   // [N,64]
    const float* Wh  = (const float*)d_in[1];   // [4,64,64]
    const float* W   = (const float*)d_in[2];   // [256,64]
    const int*   src = (const int*)d_in[3];     // [E]
    const int*   dst = (const int*)d_in[4];     // [E]
    int n = in_sizes[0] / DFEAT;
    int E = in_sizes[3];
    float* outp = (float*)d_out;

    // workspace carve-up (floats)
    float* ws = (float*)d_ws;
    size_t scN = (size_t)n * NHEAD;
    size_t o1 = (scN + 1023) & ~(size_t)1023;       // partial reductions
    size_t o2 = o1 + 1024;                          // stats (max4, sum4)
    size_t o3 = o2 + 1024;                          // neigh
    float* scores = ws;
    float4* partial4 = (float4*)(ws + o1);
    float4* stats4 = (float4*)(ws + o2);
    float* neigh = ws + o3;

    long neighN = (long)n * HD;
    k_zero4<<<4096, 256, 0, stream>>>((float4*)neigh, neighN / 4);
    k_scores<<<(n + 63) / 64, 256, 0, stream>>>(x, Wh, scores, n);
    k_pmax<<<RB, 256, 0, stream>>>((const float4*)scores, partial4, n);
    k_fmax<<<1, 256, 0, stream>>>(partial4, stats4, RB);
    k_psum<<<RB, 256, 0, stream>>>((const float4*)scores, stats4, partial4, n);
    k_fsum<<<1, 256, 0, stream>>>(partial4, stats4, RB);
    k_attn<<<RB, 256, 0, stream>>>((float4*)scores, stats4, n);
    k_edges<<<(E + 31) / 32, 256, 0, stream>>>(x, scores, src, dst, neigh, E);
    k_out<<<(n + 15) / 16, 128, 0, stream>>>(neigh, W, outp, n);
}